// YoloLoss_8830452761130
// MI455X (gfx1250) — compile-verified
//
#include <hip/hip_runtime.h>
#include <hip/hip_bf16.h>

typedef float v2f __attribute__((ext_vector_type(2)));
typedef float v8f __attribute__((ext_vector_type(8)));

#define SS 7
#define FEAT 30
#define STEPF (1.0f / 7.0f)

// Exact f32 wave(32) reduction using V_WMMA_F32_16X16X4_F32 with B == ones:
// D[m][n] = sum_k A[m][k] + C[m][n].
// A layout (32-bit, 16x4): VGPR0 lanes 0-15 -> (M=lane,K=0), lanes 16-31 -> (M=lane-16,K=2)
//                          VGPR1 lanes 0-15 -> (M=lane,K=1), lanes 16-31 -> (M=lane-16,K=3)
// So putting each lane's partial into A.x (A.y = 0) places all 32 values in
// distinct (M,K) slots; row-sums partition them exactly once.
// D layout: VGPR r, lanes 0-15 -> D[M=r][N=lane]; lanes 16-31 -> D[M=8+r][N=lane-16].
__device__ __forceinline__ float wave_reduce_wmma(float x) {
    v2f a; a.x = x;     a.y = 0.0f;
    v2f b; b.x = 1.0f;  b.y = 1.0f;
    v8f c = {0.0f, 0.0f, 0.0f, 0.0f, 0.0f, 0.0f, 0.0f, 0.0f};
    v8f d = __builtin_amdgcn_wmma_f32_16x16x4_f32(false, a, false, b, (short)0, c,
                                                  false, false);
    // lanes 0-15 hold rows 0..7 in components 0..7; lanes 16-31 hold rows 8..15
    float s = d[0] + d[1] + d[2] + d[3] + d[4] + d[5] + d[6] + d[7];
    s += __shfl_xor(s, 16, 32);   // combine the two half-row sums -> full wave sum
    return s;
}

__global__ void yolo_partial(const float* __restrict__ pred,
                             const float* __restrict__ target,
                             float* __restrict__ partial,
                             int n_cells) {
    const int tid = blockIdx.x * blockDim.x + threadIdx.x;
    const int c = (tid < n_cells) ? tid : (n_cells - 1);
    const float valid = (tid < n_cells) ? 1.0f : 0.0f;

    // 120-byte cell records are 8-byte aligned -> widest always-legal access is B64.
    const float2* p2 = reinterpret_cast<const float2*>(pred + (size_t)c * FEAT);
    const float2* t2 = reinterpret_cast<const float2*>(target + (size_t)c * FEAT);
    float p[FEAT], t[FEAT];
#pragma unroll
    for (int k = 0; k < FEAT / 2; ++k) {
        float2 a = p2[k]; p[2 * k] = a.x; p[2 * k + 1] = a.y;
        float2 b = t2[k]; t[2 * k] = b.x; t[2 * k + 1] = b.y;
    }

    const int j = c % SS;             // gx varies along innermost spatial axis
    const int i = (c / SS) % SS;      // gy along the outer spatial axis
    const float gx = (float)j, gy = (float)i;

    const float cell_mask = (t[9] > 0.0f) ? 1.0f : 0.0f;

    float iou[2];
#pragma unroll
    for (int b = 0; b < 2; ++b) {
        const int o = 5 * b;
        // _convert_boxes: clip() applies to x0,y0,w,h (all four!)
        float px0 = fmaxf((p[o + 0] + gx) * STEPF - 0.5f * p[o + 2], 0.0f);
        float py0 = fmaxf((p[o + 1] + gy) * STEPF - 0.5f * p[o + 3], 0.0f);
        float pw  = fmaxf(p[o + 2], 0.0f);
        float ph  = fmaxf(p[o + 3], 0.0f);
        float tx0 = fmaxf((t[o + 0] + gx) * STEPF - 0.5f * t[o + 2], 0.0f);
        float ty0 = fmaxf((t[o + 1] + gy) * STEPF - 0.5f * t[o + 3], 0.0f);
        float tw  = fmaxf(t[o + 2], 0.0f);
        float th  = fmaxf(t[o + 3], 0.0f);
        float iw = fmaxf(pw + tw - (fmaxf(px0 + pw, tx0 + tw) - fminf(px0, tx0)), 0.0f);
        float ih = fmaxf(ph + th - (fmaxf(py0 + ph, ty0 + th) - fminf(py0, ty0)), 0.0f);
        float inter = iw * ih;
        float uni = pw * ph + tw * th - inter;
        iou[b] = inter / (uni + 1e-10f);
    }

    // argmax over B=2 with first-max tie-break: iou0 >= iou1 -> box 0 responsible
    const float r0 = (iou[0] >= iou[1]) ? 1.0f : 0.0f;
    const float resp[2] = {r0, 1.0f - r0};

    float loss = 0.0f;
#pragma unroll
    for (int b = 0; b < 2; ++b) {
        const int o = 5 * b;
        const float ob = cell_mask * resp[b];
        const float dx = p[o + 0] - t[o + 0];
        const float dy = p[o + 1] - t[o + 1];
        const float dw = p[o + 2] - t[o + 2];
        const float dh = p[o + 3] - t[o + 3];
        loss += 5.0f * ob * (dx * dx + dy * dy + dw * dw + dh * dh);   // coord
        const float dc = p[o + 4] - iou[b];
        loss += ob * dc * dc;                                          // conf
        loss += 0.5f * (1.0f - ob) * p[o + 4] * p[o + 4];              // noobj
    }
    float cl = 0.0f;
#pragma unroll
    for (int k = 10; k < FEAT; ++k) {
        const float d = p[k] - t[k];
        cl += d * d;
    }
    loss += cell_mask * cl;                                            // class
    loss *= valid;

    // wave reduce (WMMA) -> cross-wave via LDS -> one partial per block
    const float wsum = wave_reduce_wmma(loss);
    __shared__ float lds[8];
    const int lane = threadIdx.x & 31;
    const int wave = threadIdx.x >> 5;
    if (lane == 0) lds[wave] = wsum;
    __syncthreads();
    if (threadIdx.x == 0) {
        float s = 0.0f;
        const int nw = blockDim.x >> 5;
        for (int w = 0; w < nw; ++w) s += lds[w];
        partial[blockIdx.x] = s;
    }
}

// Single-wave final reduction: chain WMMA as an accumulator (D = A*ones + C).
// Each iteration folds 64 block-partials into the 16 row accumulators exactly
// once each (unique (M,K) slot per loaded value). Deterministic order.
__global__ void yolo_final(const float* __restrict__ partial,
                           float* __restrict__ out,
                           int n_partials, float inv_n) {
    const int lane = threadIdx.x;   // launched with exactly 32 threads
    v2f ones; ones.x = 1.0f; ones.y = 1.0f;
    v8f acc = {0.0f, 0.0f, 0.0f, 0.0f, 0.0f, 0.0f, 0.0f, 0.0f};
    const int iters = (n_partials + 63) >> 6;
    for (int it = 0; it < iters; ++it) {
        const int i0 = it * 64 + lane;
        const int i1 = i0 + 32;
        v2f a;
        a.x = (i0 < n_partials) ? partial[i0] : 0.0f;
        a.y = (i1 < n_partials) ? partial[i1] : 0.0f;
        acc = __builtin_amdgcn_wmma_f32_16x16x4_f32(false, a, false, ones, (short)0,
                                                    acc, false, false);
    }
    float s = acc[0] + acc[1] + acc[2] + acc[3] + acc[4] + acc[5] + acc[6] + acc[7];
    s += __shfl_xor(s, 16, 32);
    if (lane == 0) out[0] = s * inv_n;
}

extern "C" void kernel_launch(void* const* d_in, const int* in_sizes, int n_in,
                              void* d_out, int out_size, void* d_ws, size_t ws_size,
                              hipStream_t stream) {
    (void)n_in; (void)out_size; (void)ws_size;
    const float* pred   = (const float*)d_in[0];
    const float* target = (const float*)d_in[1];
    // in_sizes[0] = N*S*S*(5*B+C); derive everything at runtime.
    const int total   = in_sizes[0];
    const int n_cells = total / FEAT;          // N*49 = 802816
    const int N       = n_cells / (SS * SS);   // 16384
    const int threads = 256;
    const int blocks  = (n_cells + threads - 1) / threads;   // 3136 (exact)

    float* partial = (float*)d_ws;             // blocks * 4 bytes of scratch

    yolo_partial<<<blocks, threads, 0, stream>>>(pred, target, partial, n_cells);
    yolo_final<<<1, 32, 0, stream>>>(partial, (float*)d_out, blocks,
                                     1.0f / (float)N);
}